// GATNet_78889959292941
// MI455X (gfx1250) — compile-verified
//
#include <hip/hip_runtime.h>
#include <hip/hip_bf16.h>
#include <math.h>

typedef __attribute__((ext_vector_type(2))) float v2f;
typedef __attribute__((ext_vector_type(8))) float v8f;

#define NEG_SLOPE 0.2f

// ---------------------------------------------------------------------------
// WMMA f32 GEMM:  Y[n,64] = X[n,K] @ W[K,64]     (K = 128 or 64)
// 128 threads = 4 wave32. Block covers 64 rows; wave w owns cols [16w,16w+16)
// and computes 4 stacked 16x16 row-tiles, so each B fragment (from W) feeds
// 4 back-to-back V_WMMA_F32_16X16X4_F32 (no D->A/B hazard between them).
// X tile (64 x K) staged in LDS (max 32 KB).
// ---------------------------------------------------------------------------
__global__ void gemm_wmma_f32(const float* __restrict__ X,
                              const float* __restrict__ W,
                              float* __restrict__ Y, int n, int K) {
  __shared__ float sA[64 * 128];
  const int row0 = blockIdx.x * 64;
  const int tid  = threadIdx.x;       // 0..127
  const int wave = tid >> 5;          // 0..3
  const int lane = tid & 31;

  // Stage 64 x K tile of X into LDS (row-clamped: no divergence, EXEC uniform).
  for (int i = tid; i < 64 * K; i += 128) {
    int r = i / K, c = i - r * K;
    int gr = row0 + r;
    if (gr >= n) gr = n - 1;
    sA[i] = X[(size_t)gr * K + c];
  }
  __syncthreads();

  const int half = lane >> 4;         // 0: lanes 0-15, 1: lanes 16-31
  const int m    = lane & 15;
  const int col  = wave * 16 + m;

  v8f acc[4] = {};
  for (int k = 0; k < K; k += 4) {
    // B 4x16 fragment: lanes 0-15 hold K=k,k+1 ; lanes 16-31 hold K=k+2,k+3
    v2f b;
    b.x = W[(size_t)(k + 2 * half) * 64 + col];
    b.y = W[(size_t)(k + 2 * half + 1) * 64 + col];
#pragma unroll
    for (int t = 0; t < 4; ++t) {
      // A 16x4 fragment for row-tile t (same K striping as B)
      v2f a;
      a.x = sA[(16 * t + m) * K + k + 2 * half];
      a.y = sA[(16 * t + m) * K + k + 2 * half + 1];
      acc[t] = __builtin_amdgcn_wmma_f32_16x16x4_f32(
          /*neg_a=*/false, a, /*neg_b=*/false, b,
          /*c_mod=*/(short)0, acc[t], /*reuse_a=*/false, /*reuse_b=*/false);
    }
  }

  // C/D layout: VGPR j -> row j (lanes 0-15) / row j+8 (lanes 16-31)
#pragma unroll
  for (int t = 0; t < 4; ++t) {
#pragma unroll
    for (int j = 0; j < 8; ++j) {
      int r = row0 + 16 * t + j + 8 * half;
      if (r < n) Y[(size_t)r * 64 + col] = acc[t][j];
    }
  }
}

// ---------------------------------------------------------------------------
// Helpers
// ---------------------------------------------------------------------------
__device__ __forceinline__ void atomicMaxF(float* addr, float val) {
  int* ai = (int*)addr;
  int old = __float_as_int(*addr);
  while (__int_as_float(old) < val) {
    int assumed = old;
    old = atomicCAS(ai, assumed, __float_as_int(val));
    if (old == assumed) break;
  }
}

__device__ __forceinline__ float leaky(float x) {
  return x > 0.0f ? x : NEG_SLOPE * x;
}
__device__ __forceinline__ float eluf(float x) {
  return x > 0.0f ? x : expm1f(x);
}

__global__ void fill_f32(float* p, float v, int count) {
  int i = blockIdx.x * blockDim.x + threadIdx.x;
  if (i < count) p[i] = v;
}

// ---------------------------------------------------------------------------
// Attention scores
// ---------------------------------------------------------------------------
// Layer 1: 8 heads x 8 ch.  thread = node*8 + head
__global__ void att_scores_h8(const float* __restrict__ h,
                              const float* __restrict__ a_src,
                              const float* __restrict__ a_dst,
                              float* __restrict__ as_, float* __restrict__ ad_,
                              int n) {
  int i = blockIdx.x * blockDim.x + threadIdx.x;
  if (i >= n * 8) return;
  int node = i >> 3, hd = i & 7;
  const float* hp = h + (size_t)node * 64 + hd * 8;
  float s = 0.0f, d = 0.0f;
#pragma unroll
  for (int c = 0; c < 8; ++c) {
    float v = hp[c];
    s += v * a_src[hd * 8 + c];
    d += v * a_dst[hd * 8 + c];
  }
  as_[i] = s;
  ad_[i] = d;
}

// Layer 2: 1 head x 64 ch.  thread = node
__global__ void att_scores_h1(const float* __restrict__ h,
                              const float* __restrict__ a_src,
                              const float* __restrict__ a_dst,
                              float* __restrict__ as_, float* __restrict__ ad_,
                              int n) {
  int node = blockIdx.x * blockDim.x + threadIdx.x;
  if (node >= n) return;
  const float* hp = h + (size_t)node * 64;
  float s = 0.0f, d = 0.0f;
#pragma unroll 8
  for (int c = 0; c < 64; ++c) {
    float v = hp[c];
    s += v * a_src[c];
    d += v * a_dst[c];
  }
  as_[node] = s;
  ad_[node] = d;
}

// ---------------------------------------------------------------------------
// Edge passes.  Edge id e in [0,E) from edge_index; e in [E,E+n) = self loop.
// ---------------------------------------------------------------------------
__device__ __forceinline__ void edge_sd(const int* ei, int E, int e, int& s, int& d) {
  if (e < E) { s = ei[e]; d = ei[E + e]; }
  else       { s = d = e - E; }
}

// ------------------------------ layer 1 (8 heads) --------------------------
__global__ void edge_max_h8(const int* __restrict__ ei, int E, int n,
                            const float* __restrict__ as_, const float* __restrict__ ad_,
                            float* __restrict__ mx) {
  int i = blockIdx.x * blockDim.x + threadIdx.x;
  int tot = (E + n) * 8;
  if (i >= tot) return;
  int e = i >> 3, hd = i & 7;
  int s, d; edge_sd(ei, E, e, s, d);
  float ev = leaky(as_[s * 8 + hd] + ad_[d * 8 + hd]);
  atomicMaxF(&mx[d * 8 + hd], ev);
}

__global__ void edge_sum_h8(const int* __restrict__ ei, int E, int n,
                            const float* __restrict__ as_, const float* __restrict__ ad_,
                            const float* __restrict__ mx, float* __restrict__ dn) {
  int i = blockIdx.x * blockDim.x + threadIdx.x;
  int tot = (E + n) * 8;
  if (i >= tot) return;
  int e = i >> 3, hd = i & 7;
  int s, d; edge_sd(ei, E, e, s, d);
  float ev = leaky(as_[s * 8 + hd] + ad_[d * 8 + hd]);
  atomicAdd(&dn[d * 8 + hd], expf(ev - mx[d * 8 + hd]));
}

// wave per edge; lane handles channels 2*lane, 2*lane+1 (contiguous float2
// gather: the wave reads one 256B line of h[src]; both channels share a head)
__global__ void edge_scatter_h8(const int* __restrict__ ei, int E, int n,
                                const float* __restrict__ as_, const float* __restrict__ ad_,
                                const float* __restrict__ mx, const float* __restrict__ dn,
                                const float* __restrict__ h, float* __restrict__ out) {
  int wid = (blockIdx.x * blockDim.x + threadIdx.x) >> 5;
  int lane = threadIdx.x & 31;
  int tot = E + n;
  if (wid >= tot) return;
  int s, d; edge_sd(ei, E, wid, s, d);
  int c0 = lane * 2;
  int hd = c0 >> 3;
  float ev = leaky(as_[s * 8 + hd] + ad_[d * 8 + hd]);
  float alpha = expf(ev - mx[d * 8 + hd]) / fmaxf(dn[d * 8 + hd], 1e-16f);
  const float2 hv = *(const float2*)(h + (size_t)s * 64 + c0);
  float* op = out + (size_t)d * 64 + c0;
  atomicAdd(op,     hv.x * alpha);
  atomicAdd(op + 1, hv.y * alpha);
}

// ------------------------------ layer 2 (1 head) ---------------------------
__global__ void edge_max_h1(const int* __restrict__ ei, int E, int n,
                            const float* __restrict__ as_, const float* __restrict__ ad_,
                            float* __restrict__ mx) {
  int e = blockIdx.x * blockDim.x + threadIdx.x;
  if (e >= E + n) return;
  int s, d; edge_sd(ei, E, e, s, d);
  atomicMaxF(&mx[d], leaky(as_[s] + ad_[d]));
}

__global__ void edge_sum_h1(const int* __restrict__ ei, int E, int n,
                            const float* __restrict__ as_, const float* __restrict__ ad_,
                            const float* __restrict__ mx, float* __restrict__ dn) {
  int e = blockIdx.x * blockDim.x + threadIdx.x;
  if (e >= E + n) return;
  int s, d; edge_sd(ei, E, e, s, d);
  float ev = leaky(as_[s] + ad_[d]);
  atomicAdd(&dn[d], expf(ev - mx[d]));
}

__global__ void edge_scatter_h1(const int* __restrict__ ei, int E, int n,
                                const float* __restrict__ as_, const float* __restrict__ ad_,
                                const float* __restrict__ mx, const float* __restrict__ dn,
                                const float* __restrict__ h, float* __restrict__ out) {
  int wid = (blockIdx.x * blockDim.x + threadIdx.x) >> 5;
  int lane = threadIdx.x & 31;
  int tot = E + n;
  if (wid >= tot) return;
  int s, d; edge_sd(ei, E, wid, s, d);
  float ev = leaky(as_[s] + ad_[d]);
  float alpha = expf(ev - mx[d]) / fmaxf(dn[d], 1e-16f);
  int c0 = lane * 2;
  const float2 hv = *(const float2*)(h + (size_t)s * 64 + c0);
  float* op = out + (size_t)d * 64 + c0;
  atomicAdd(op,     hv.x * alpha);
  atomicAdd(op + 1, hv.y * alpha);
}

// ---------------------------------------------------------------------------
// ELU(out + bias)  ->  dst    (layer-1 activation)
// ---------------------------------------------------------------------------
__global__ void elu_bias(const float* __restrict__ src, const float* __restrict__ b,
                         float* __restrict__ dst, int n) {
  int i = blockIdx.x * blockDim.x + threadIdx.x;
  if (i >= n * 64) return;
  dst[i] = eluf(src[i] + b[i & 63]);
}

// ---------------------------------------------------------------------------
// global mean pool: scatter sums + counts
// ---------------------------------------------------------------------------
__global__ void pool_scatter(const float* __restrict__ h, const float* __restrict__ b2,
                             const int* __restrict__ batch,
                             float* __restrict__ pooled, float* __restrict__ cnt, int n) {
  int i = blockIdx.x * blockDim.x + threadIdx.x;
  if (i >= n * 64) return;
  int node = i >> 6, c = i & 63;
  int g = batch[node];
  atomicAdd(&pooled[(size_t)g * 64 + c], h[i] + b2[c]);
  if (c == 0) atomicAdd(&cnt[g], 1.0f);
}

// g1 = elu( (pooled/cnt) @ lw1 + lb1 )      thread = g*128 + j
__global__ void mlp1(const float* __restrict__ pooled, const float* __restrict__ cnt,
                     const float* __restrict__ lw1, const float* __restrict__ lb1,
                     float* __restrict__ g1, int G) {
  int i = blockIdx.x * blockDim.x + threadIdx.x;
  if (i >= G * 128) return;
  int g = i >> 7, j = i & 127;
  float inv = 1.0f / fmaxf(cnt[g], 1.0f);
  float acc = lb1[j];
#pragma unroll 8
  for (int k = 0; k < 64; ++k)
    acc += (pooled[g * 64 + k] * inv) * lw1[k * 128 + j];
  g1[i] = eluf(acc);
}

// out = g1 @ lw2 + lb2       thread = g
__global__ void mlp2(const float* __restrict__ g1, const float* __restrict__ lw2,
                     const float* __restrict__ lb2, float* __restrict__ out, int G) {
  int g = blockIdx.x * blockDim.x + threadIdx.x;
  if (g >= G) return;
  float acc = lb2[0];
#pragma unroll 8
  for (int k = 0; k < 128; ++k) acc += g1[g * 128 + k] * lw2[k];
  out[g] = acc;
}

// ---------------------------------------------------------------------------
// Host-side orchestration
// ---------------------------------------------------------------------------
extern "C" void kernel_launch(void* const* d_in, const int* in_sizes, int n_in,
                              void* d_out, int out_size, void* d_ws, size_t ws_size,
                              hipStream_t stream) {
  const float* x        = (const float*)d_in[0];
  const int*   ei       = (const int*)  d_in[1];
  const int*   batch    = (const int*)  d_in[2];
  const float* W1       = (const float*)d_in[3];
  const float* att_src1 = (const float*)d_in[4];
  const float* att_dst1 = (const float*)d_in[5];
  const float* b1       = (const float*)d_in[6];
  const float* W2       = (const float*)d_in[7];
  const float* att_src2 = (const float*)d_in[8];
  const float* att_dst2 = (const float*)d_in[9];
  const float* b2       = (const float*)d_in[10];
  const float* lw1      = (const float*)d_in[11];
  const float* lb1      = (const float*)d_in[12];
  const float* lw2      = (const float*)d_in[13];
  const float* lb2      = (const float*)d_in[14];
  float*       out      = (float*)d_out;

  const int n = in_sizes[0] / 128;   // nodes
  const int E = in_sizes[1] / 2;     // edges (before self loops)
  const int G = out_size;            // graphs
  const int tot = E + n;             // edges incl. self loops

  // workspace layout (f32)
  float* bufA   = (float*)d_ws;                 // n*64 : h1 -> hE -> out2
  float* bufB   = bufA + (size_t)n * 64;        // n*64 : out1 -> h2
  float* as1    = bufB + (size_t)n * 64;        // n*8
  float* ad1    = as1 + (size_t)n * 8;
  float* m1     = ad1 + (size_t)n * 8;
  float* d1     = m1  + (size_t)n * 8;
  float* as2    = d1  + (size_t)n * 8;          // n
  float* ad2    = as2 + n;
  float* m2     = ad2 + n;
  float* d2     = m2  + n;
  float* pooled = d2  + n;                      // G*64
  float* cnt    = pooled + (size_t)G * 64;      // G
  float* g1     = cnt + G;                      // G*128

  const int B = 256;
  auto blocks = [](long long w, int b) { return (int)((w + b - 1) / b); };

  // ---------------- layer 1 ----------------
  gemm_wmma_f32<<<(n + 63) / 64, 128, 0, stream>>>(x, W1, bufA, n, 128);
  att_scores_h8<<<blocks((long long)n * 8, B), B, 0, stream>>>(bufA, att_src1, att_dst1, as1, ad1, n);

  fill_f32<<<blocks((long long)n * 8, B), B, 0, stream>>>(m1, -1e30f, n * 8);
  hipMemsetAsync(d1, 0, (size_t)n * 8 * sizeof(float), stream);
  hipMemsetAsync(bufB, 0, (size_t)n * 64 * sizeof(float), stream);

  edge_max_h8<<<blocks((long long)tot * 8, B), B, 0, stream>>>(ei, E, n, as1, ad1, m1);
  edge_sum_h8<<<blocks((long long)tot * 8, B), B, 0, stream>>>(ei, E, n, as1, ad1, m1, d1);
  edge_scatter_h8<<<blocks((long long)tot * 32, B), B, 0, stream>>>(ei, E, n, as1, ad1, m1, d1, bufA, bufB);

  // hE = elu(out1 + b1)  (overwrite bufA)
  elu_bias<<<blocks((long long)n * 64, B), B, 0, stream>>>(bufB, b1, bufA, n);

  // ---------------- layer 2 ----------------
  gemm_wmma_f32<<<(n + 63) / 64, 128, 0, stream>>>(bufA, W2, bufB, n, 64);
  att_scores_h1<<<blocks(n, B), B, 0, stream>>>(bufB, att_src2, att_dst2, as2, ad2, n);

  fill_f32<<<blocks(n, B), B, 0, stream>>>(m2, -1e30f, n);
  hipMemsetAsync(d2, 0, (size_t)n * sizeof(float), stream);
  hipMemsetAsync(bufA, 0, (size_t)n * 64 * sizeof(float), stream);

  edge_max_h1<<<blocks(tot, B), B, 0, stream>>>(ei, E, n, as2, ad2, m2);
  edge_sum_h1<<<blocks(tot, B), B, 0, stream>>>(ei, E, n, as2, ad2, m2, d2);
  edge_scatter_h1<<<blocks((long long)tot * 32, B), B, 0, stream>>>(ei, E, n, as2, ad2, m2, d2, bufB, bufA);

  // ---------------- pool + MLP ----------------
  hipMemsetAsync(pooled, 0, (size_t)G * 64 * sizeof(float), stream);
  hipMemsetAsync(cnt, 0, (size_t)G * sizeof(float), stream);
  pool_scatter<<<blocks((long long)n * 64, B), B, 0, stream>>>(bufA, b2, batch, pooled, cnt, n);

  mlp1<<<blocks((long long)G * 128, B), B, 0, stream>>>(pooled, cnt, lw1, lb1, g1, G);
  mlp2<<<blocks(G, B), B, 0, stream>>>(g1, lw2, lb2, out, G);
}